// HypRegressionPoincareHDRL_8160437862874
// MI455X (gfx1250) — compile-verified
//
#include <hip/hip_runtime.h>
#include <math.h>

// ---------------------------------------------------------------------------
// Poincare MLR (HDRL) fused kernel set for gfx1250 (MI455X).
//   B=8192, D_IN=1024, D_OUT=4096, c=1.0
//   p = s * bias (row scalar s from Poincare projection)
//   px = s_n * (x @ bias^T), xa = x @ weight^T  -> fused asinh epilogue
// Dual-GEMM on V_WMMA_F32_16X16X4_F32; async global->LDS double buffering.
// ---------------------------------------------------------------------------

typedef __attribute__((ext_vector_type(2))) float v2f;
typedef __attribute__((ext_vector_type(4))) float f4;
typedef __attribute__((ext_vector_type(8))) float v8f;
typedef __attribute__((ext_vector_type(4))) int   v4i;

#define BATCH   8192
#define DIN     1024
#define DOUT    4096
#define CCURV   1.0f
#define PROJEPS 1e-5f

// block tile 128x128, K-slab 16, double buffered
#define MT   128
#define NT   128
#define KC   16
#define LSTR 20   // padded LDS row stride (dwords): rows 16B-aligned, b64 frags bank-conflict-free

// ---- gfx1250 async global->LDS path (guarded; falls back to sync copy) ----
#if __has_builtin(__builtin_amdgcn_global_load_async_to_lds_b128)
#define USE_ASYNC_LDS 1
#else
#define USE_ASYNC_LDS 0
#endif

#if USE_ASYNC_LDS
__device__ __forceinline__ void cp16(void* lds, const void* g) {
  // generic->AS1/AS3 via integer casts (flat addr low 32 bits == LDS offset)
  __builtin_amdgcn_global_load_async_to_lds_b128(
      (__attribute__((address_space(1))) v4i*)(uintptr_t)g,
      (__attribute__((address_space(3))) v4i*)(uint32_t)(uintptr_t)lds,
      0, 0);
}
#if __has_builtin(__builtin_amdgcn_s_wait_asynccnt)
#define WAIT_ASYNC(n) __builtin_amdgcn_s_wait_asynccnt(n)
#else
#define WAIT_ASYNC(n) asm volatile("s_wait_asynccnt %0" ::"i"(n))
#endif
#else
__device__ __forceinline__ void cp16(void* lds, const void* g) {
  *(f4*)lds = *(const f4*)g;
}
#define WAIT_ASYNC(n)
#endif

__device__ __forceinline__ float waveRed(float v) {
#pragma unroll
  for (int m = 16; m >= 1; m >>= 1) v += __shfl_xor(v, m, 32);
  return v;
}

// ---------------------------------------------------------------------------
// Per-output-row stats: scale s (Poincare projection), pa=<p,a>, p2=|p|^2, |a|
// ---------------------------------------------------------------------------
__global__ __launch_bounds__(256) void row_stats_kernel(
    const float* __restrict__ weight, const float* __restrict__ bias,
    float* __restrict__ scale, float* __restrict__ pa,
    float* __restrict__ p2, float* __restrict__ an) {
  const int row  = (int)((blockIdx.x * blockDim.x + threadIdx.x) >> 5);
  const int lane = (int)(threadIdx.x & 31);
  if (row >= DOUT) return;
  const float* b = bias   + (size_t)row * DIN;
  const float* w = weight + (size_t)row * DIN;
  float b2 = 0.f, w2 = 0.f, bw = 0.f;
  for (int j = lane; j < DIN; j += 32) {
    float bv = b[j], wv = w[j];
    b2 = fmaf(bv, bv, b2);
    w2 = fmaf(wv, wv, w2);
    bw = fmaf(bv, wv, bw);
  }
  b2 = waveRed(b2); w2 = waveRed(w2); bw = waveRed(bw);
  if (lane == 0) {
    float maxn = (1.0f - PROJEPS) / sqrtf(CCURV);
    float nb   = sqrtf(b2);
    float s    = (nb > maxn) ? (maxn / nb) : 1.0f;
    scale[row] = s;
    pa[row]    = s * bw;
    p2[row]    = s * s * b2;
    an[row]    = sqrtf(w2);
  }
}

// ---------------------------------------------------------------------------
// Per-batch-row |x|^2
// ---------------------------------------------------------------------------
__global__ __launch_bounds__(256) void x2_kernel(const float* __restrict__ x,
                                                 float* __restrict__ x2) {
  const int row  = (int)((blockIdx.x * blockDim.x + threadIdx.x) >> 5);
  const int lane = (int)(threadIdx.x & 31);
  if (row >= BATCH) return;
  const float* xr = x + (size_t)row * DIN;
  float s = 0.f;
  for (int j = lane; j < DIN; j += 32) { float v = xr[j]; s = fmaf(v, v, s); }
  s = waveRed(s);
  if (lane == 0) x2[row] = s;
}

// ---------------------------------------------------------------------------
// One K-slab of tile loads: each of 256 threads moves 6 x 16B (A/P/W tiles).
// lr = tid>>2 (row 0..63), lc = (tid&3)*4 (dword col).
// ---------------------------------------------------------------------------
__device__ __forceinline__ void copy_phase(const float* __restrict__ gA,
                                           const float* __restrict__ gP,
                                           const float* __restrict__ gW,
                                           float* sA, float* sP, float* sW,
                                           int lr, int lc) {
#pragma unroll
  for (int ps = 0; ps < 2; ++ps) {
    const int r = lr + ps * 64;
    cp16(sA + r * LSTR + lc, gA + (size_t)r * DIN + lc);
    cp16(sP + r * LSTR + lc, gP + (size_t)r * DIN + lc);
    cp16(sW + r * LSTR + lc, gW + (size_t)r * DIN + lc);
  }
}

// ---------------------------------------------------------------------------
// Fused dual-GEMM + Poincare MLR epilogue.
// Block: 256 thr (8 waves), tile 128x128. Wave grid 2(M)x4(N);
// each wave owns 4x2 16x16 tiles x 2 accumulators (px, xa) = 128 acc VGPRs.
// ---------------------------------------------------------------------------
__global__ __launch_bounds__(256) void pmlr_fused_kernel(
    const float* __restrict__ x, const float* __restrict__ weight,
    const float* __restrict__ bias, const float* __restrict__ scale,
    const float* __restrict__ pa, const float* __restrict__ p2,
    const float* __restrict__ an, const float* __restrict__ x2,
    float* __restrict__ out) {
  __shared__ float As[2][MT * LSTR];  // x tile       128 x 16 (x2 buffers)
  __shared__ float Ps[2][NT * LSTR];  // bias tile    128 x 16
  __shared__ float Ws[2][NT * LSTR];  // weight tile  128 x 16

  const int tid   = (int)threadIdx.x;
  const int lane  = tid & 31;
  const int wv    = tid >> 5;
  const int waveM = wv & 1;    // 0..1 -> 64-row slab
  const int waveN = wv >> 1;   // 0..3 -> 32-col slab
  const int l15   = lane & 15;
  const int kl    = lane >> 4; // hi half-wave handles K+2..K+3

  const int mBlock = (int)blockIdx.y * MT;
  const int nBlock = (int)blockIdx.x * NT;

  const float* gA = x      + (size_t)mBlock * DIN;
  const float* gP = bias   + (size_t)nBlock * DIN;
  const float* gW = weight + (size_t)nBlock * DIN;

  const int lr = tid >> 2;        // 0..63
  const int lc = (tid & 3) << 2;  // 0,4,8,12

  v8f accP[4][2], accA[4][2];
  const v8f zero8 = {0.f, 0.f, 0.f, 0.f, 0.f, 0.f, 0.f, 0.f};
#pragma unroll
  for (int i = 0; i < 4; ++i)
#pragma unroll
    for (int j = 0; j < 2; ++j) { accP[i][j] = zero8; accA[i][j] = zero8; }

  // prologue: stage phase 0 into buffer 0
  copy_phase(gA, gP, gW, As[0], Ps[0], Ws[0], lr, lc);

  const int NPH = DIN / KC;  // 64
  for (int ph = 0; ph < NPH; ++ph) {
    const int buf = ph & 1;
    if (ph + 1 < NPH) {
      const int kb = (ph + 1) * KC;
      copy_phase(gA + kb, gP + kb, gW + kb, As[buf ^ 1], Ps[buf ^ 1],
                 Ws[buf ^ 1], lr, lc);
      WAIT_ASYNC(6);  // allow next slab's 6 loads in flight; current slab done
    } else {
      WAIT_ASYNC(0);
    }
    __syncthreads();

    const float* bA = As[buf];
    const float* bP = Ps[buf];
    const float* bW = Ws[buf];
#pragma unroll
    for (int kk = 0; kk < KC; kk += 4) {
      v2f af[4], pf[2], wf[2];
#pragma unroll
      for (int i = 0; i < 4; ++i) {  // A frag: M=l15(+tile), K=kk+kl*2+{0,1}
        const int row = waveM * 64 + i * 16 + l15;
        af[i] = *(const v2f*)(bA + row * LSTR + kk + kl * 2);
      }
#pragma unroll
      for (int j = 0; j < 2; ++j) {  // B frags (N x K row-major == B^T)
        const int row = waveN * 32 + j * 16 + l15;
        pf[j] = *(const v2f*)(bP + row * LSTR + kk + kl * 2);
        wf[j] = *(const v2f*)(bW + row * LSTR + kk + kl * 2);
      }
#pragma unroll
      for (int i = 0; i < 4; ++i)
#pragma unroll
        for (int j = 0; j < 2; ++j) {
          accP[i][j] = __builtin_amdgcn_wmma_f32_16x16x4_f32(
              false, af[i], false, pf[j], (short)0, accP[i][j], false, false);
          accA[i][j] = __builtin_amdgcn_wmma_f32_16x16x4_f32(
              false, af[i], false, wf[j], (short)0, accA[i][j], false, false);
        }
    }
    __syncthreads();
  }

  // ---------------- epilogue: Mobius-add distance, asinh ----------------
  const float sqc = sqrtf(CCURV);
#pragma unroll
  for (int j = 0; j < 2; ++j) {
    const int n = nBlock + waveN * 32 + j * 16 + l15;
    const float sc_n = scale[n];
    const float pa_n = pa[n];
    const float p2_n = p2[n];
    const float an_n = an[n];
    const float Bc = 1.0f - CCURV * p2_n;
#pragma unroll
    for (int i = 0; i < 4; ++i) {
      const int mBase = mBlock + waveM * 64 + i * 16 + kl * 8;
#pragma unroll
      for (int v = 0; v < 8; ++v) {
        const int m = mBase + v;
        const float x2m = x2[m];
        const float PX = sc_n * accP[i][j][v];  // <p_n, x_m>
        const float XA = accA[i][j][v];         // <x_m, a_n>
        const float A_ = 1.0f - 2.0f * CCURV * PX + CCURV * x2m;
        const float Dd = 1.0f - 2.0f * CCURV * PX + (CCURV * CCURV) * p2_n * x2m;
        const float invD = 1.0f / Dd;
        float dn2 = (A_ * A_ * p2_n - 2.0f * A_ * Bc * PX + Bc * Bc * x2m) * invD * invD;
        dn2 = fmaxf(dn2, 1e-15f);
        const float num = 2.0f * (Bc * XA - A_ * pa_n) * invD;
        const float den = fabsf((1.0f - CCURV * dn2) * an_n) + 1e-15f;
        const float z = sqc * num / den;
        out[(size_t)m * DOUT + n] = (asinhf(z) / sqc) * an_n;
      }
    }
  }
}

// ---------------------------------------------------------------------------
extern "C" void kernel_launch(void* const* d_in, const int* in_sizes, int n_in,
                              void* d_out, int out_size, void* d_ws, size_t ws_size,
                              hipStream_t stream) {
  (void)in_sizes; (void)n_in; (void)out_size; (void)ws_size;
  const float* x      = (const float*)d_in[0];
  const float* weight = (const float*)d_in[1];
  const float* bias   = (const float*)d_in[2];
  float* out = (float*)d_out;

  float* ws    = (float*)d_ws;  // 4*DOUT + BATCH floats = 96 KB scratch
  float* scale = ws;
  float* pa    = ws + DOUT;
  float* p2    = ws + 2 * DOUT;
  float* an    = ws + 3 * DOUT;
  float* x2    = ws + 4 * DOUT;

  row_stats_kernel<<<DOUT / 8, 256, 0, stream>>>(weight, bias, scale, pa, p2, an);
  x2_kernel<<<BATCH / 8, 256, 0, stream>>>(x, x2);

  dim3 grid(DOUT / NT, BATCH / MT);  // 32 x 64 blocks
  pmlr_fused_kernel<<<grid, 256, 0, stream>>>(x, weight, bias, scale, pa, p2, an,
                                              x2, out);
}